// SimpleGCNEncoder_91079076479585
// MI455X (gfx1250) — compile-verified
//
#include <hip/hip_runtime.h>

typedef float v2f __attribute__((ext_vector_type(2)));
typedef float v8f __attribute__((ext_vector_type(8)));

#define FDIM 128

// ---------------- CSR-by-destination construction ----------------

__global__ __launch_bounds__(256) void zero_counts_kernel(unsigned* __restrict__ counts, int n) {
    int i = blockIdx.x * 256 + threadIdx.x;
    if (i < n) counts[i] = 0u;
}

__global__ __launch_bounds__(256) void count_edges_kernel(const long long* __restrict__ dst,
                                                          unsigned* __restrict__ counts, int e) {
    int i = blockIdx.x * 256 + threadIdx.x;
    if (i < e) atomicAdd(&counts[dst[i]], 1u);
}

// Single-workgroup chunked exclusive scan: rowptr[0..n] from counts[0..n).
__global__ __launch_bounds__(256) void scan_kernel(const unsigned* __restrict__ counts,
                                                   unsigned* __restrict__ rowptr, int n) {
    __shared__ unsigned sbuf[256];
    __shared__ unsigned carry;
    if (threadIdx.x == 0) carry = 0u;
    __syncthreads();
    for (int base = 0; base < n; base += 256) {
        int i = base + (int)threadIdx.x;
        unsigned v = (i < n) ? counts[i] : 0u;
        unsigned c = carry;
        sbuf[threadIdx.x] = v;
        __syncthreads();
#pragma unroll
        for (int off = 1; off < 256; off <<= 1) {
            unsigned t = (threadIdx.x >= (unsigned)off) ? sbuf[threadIdx.x - off] : 0u;
            __syncthreads();
            sbuf[threadIdx.x] += t;
            __syncthreads();
        }
        if (i < n) rowptr[i] = c + sbuf[threadIdx.x] - v;   // exclusive prefix
        unsigned total = sbuf[255];
        __syncthreads();
        if (threadIdx.x == 0) carry = c + total;
        __syncthreads();
    }
    if (threadIdx.x == 0) rowptr[n] = carry;
}

// cursor = rowptr copy; dinv = rsqrt(deg) with deg = in-edge count + 1 (self loop)
__global__ __launch_bounds__(256) void prep_kernel(const unsigned* __restrict__ rowptr,
                                                   const unsigned* __restrict__ counts,
                                                   unsigned* __restrict__ cursor,
                                                   float* __restrict__ dinv, int n) {
    int i = blockIdx.x * 256 + threadIdx.x;
    if (i < n) {
        cursor[i] = rowptr[i];
        dinv[i] = rsqrtf((float)counts[i] + 1.0f);
    }
}

__global__ __launch_bounds__(256) void fill_csr_kernel(const long long* __restrict__ src,
                                                       const long long* __restrict__ dst,
                                                       unsigned* __restrict__ cursor,
                                                       int* __restrict__ colidx, int e) {
    int i = blockIdx.x * 256 + threadIdx.x;
    if (i < e) {
        unsigned pos = atomicAdd(&cursor[dst[i]], 1u);
        colidx[pos] = (int)src[i];
    }
}

// ---------------- WMMA fp32 GEMM ----------------
// One workgroup (8 waves) computes a 16-row x 128-col output block; wave w owns
// the 16x16 tile at columns [16w,16w+16). V_WMMA_F32_16X16X4_F32, K=4/issue.
// LAYER path: out = (X @ W) * dinv[row]                  (P, gathered later)
// K path:     out = (1 - alpha) * (X @ W + bias[col])    (blend pre-term)
template <int KD, bool LAYER>
__global__ __launch_bounds__(256) void gemm_wmma_kernel(
    const float* __restrict__ X,
    const float* __restrict__ W,
    const float* __restrict__ dinv,
    const float* __restrict__ bias,
    const float* __restrict__ alphaPtr,
    float* __restrict__ out,
    int nRows)
{
    __shared__ float Xs[16 * KD];
    const int row0 = blockIdx.x * 16;

    for (int i = threadIdx.x; i < 16 * KD; i += 256) {
        int r = row0 + (i / KD);
        Xs[i] = (r < nRows) ? X[(size_t)r * KD + (i % KD)] : 0.0f;
    }
    __syncthreads();

    const int wave = threadIdx.x >> 5;
    const int lane = threadIdx.x & 31;
    const int half = lane >> 4;
    const int lm   = lane & 15;
    const int col0 = wave * 16;

    v8f acc = {};
#pragma unroll
    for (int k0 = 0; k0 < KD; k0 += 4) {
        v2f a, b;
        a.x = Xs[lm * KD + k0 + 2 * half + 0];
        a.y = Xs[lm * KD + k0 + 2 * half + 1];
        b.x = W[(size_t)(k0 + 2 * half + 0) * FDIM + col0 + lm];
        b.y = W[(size_t)(k0 + 2 * half + 1) * FDIM + col0 + lm];
        acc = __builtin_amdgcn_wmma_f32_16x16x4_f32(
            false, a, false, b, (short)0, acc, false, false);
    }

    const int col = col0 + lm;
    if (LAYER) {
#pragma unroll
        for (int j = 0; j < 8; ++j) {
            int r = row0 + j + 8 * half;
            if (r < nRows) out[(size_t)r * FDIM + col] = acc[j] * dinv[r];
        }
    } else {
        float oma = 1.0f - alphaPtr[0];
        float bb  = bias[col];
#pragma unroll
        for (int j = 0; j < 8; ++j) {
            int r = row0 + j + 8 * half;
            if (r < nRows) out[(size_t)r * FDIM + col] = oma * (acc[j] + bb);
        }
    }
}

// ---------------- gather-only aggregation, fused epilogue ----------------
// One wave per destination node; lane l owns floats [4l,4l+4). Accumulates
// self-loop + all in-neighbors of P in registers (no float atomics), then:
//   RELU path:  out[i] = relu(dinv[i]*sum + b[col])          (activations)
//   BLEND path: out[i] = alpha*(dinv[i]*sum + b[col]) + out[i] (out holds (1-a)k)
template <bool RELU>
__global__ __launch_bounds__(256) void aggregate_kernel(
    const float* __restrict__ P,
    const unsigned* __restrict__ rowptr,
    const int* __restrict__ colidx,
    const float* __restrict__ dinv,
    const float* __restrict__ bias,
    const float* __restrict__ alphaPtr,
    float* __restrict__ out,
    int n)
{
    int node = (int)((blockIdx.x * 256u + threadIdx.x) >> 5);
    int lane = threadIdx.x & 31;
    if (node >= n) return;
    const int c = lane * 4;

    float4 acc = *reinterpret_cast<const float4*>(P + (size_t)node * FDIM + c); // self loop
    unsigned j   = rowptr[node];
    unsigned end = rowptr[node + 1];
    for (; j + 1 < end; j += 2) {          // 2-way unroll for memory-level parallelism
        int s0 = colidx[j];
        int s1 = colidx[j + 1];
        float4 p0 = *reinterpret_cast<const float4*>(P + (size_t)s0 * FDIM + c);
        float4 p1 = *reinterpret_cast<const float4*>(P + (size_t)s1 * FDIM + c);
        acc.x += p0.x + p1.x;  acc.y += p0.y + p1.y;
        acc.z += p0.z + p1.z;  acc.w += p0.w + p1.w;
    }
    if (j < end) {
        int s0 = colidx[j];
        float4 p0 = *reinterpret_cast<const float4*>(P + (size_t)s0 * FDIM + c);
        acc.x += p0.x;  acc.y += p0.y;  acc.z += p0.z;  acc.w += p0.w;
    }

    float di = dinv[node];
    float4 b4 = *reinterpret_cast<const float4*>(bias + c);
    float4 r;
    r.x = di * acc.x + b4.x;  r.y = di * acc.y + b4.y;
    r.z = di * acc.z + b4.z;  r.w = di * acc.w + b4.w;

    float* op = out + (size_t)node * FDIM + c;
    if (RELU) {
        r.x = fmaxf(r.x, 0.0f);  r.y = fmaxf(r.y, 0.0f);
        r.z = fmaxf(r.z, 0.0f);  r.w = fmaxf(r.w, 0.0f);
        *reinterpret_cast<float4*>(op) = r;
    } else {
        float a = alphaPtr[0];
        float4 o = *reinterpret_cast<const float4*>(op);
        o.x = a * r.x + o.x;  o.y = a * r.y + o.y;
        o.z = a * r.z + o.z;  o.w = a * r.w + o.w;
        *reinterpret_cast<float4*>(op) = o;
    }
}

// ---------------- driver ----------------

extern "C" void kernel_launch(void* const* d_in, const int* in_sizes, int n_in,
                              void* d_out, int out_size, void* d_ws, size_t ws_size,
                              hipStream_t stream) {
    const float*     X     = (const float*)d_in[0];       // [N,128]
    const float*     Kp    = (const float*)d_in[1];       // [N,64]
    const long long* ei    = (const long long*)d_in[2];   // [2,E] int64
    const float*     W1    = (const float*)d_in[3];
    const float*     b1    = (const float*)d_in[4];
    const float*     W2    = (const float*)d_in[5];
    const float*     b2    = (const float*)d_in[6];
    const float*     kW    = (const float*)d_in[7];       // [64,128]
    const float*     kb    = (const float*)d_in[8];
    const float*     alpha = (const float*)d_in[9];

    const int n = in_sizes[0] / FDIM;
    const int e = in_sizes[2] / 2;
    float* out = (float*)d_out;

    // workspace: dinv[N] | P[N,128] | T[N,128] | counts[N] | rowptr[N+1] | cursor[N] | colidx[E]
    float*    dinv   = (float*)d_ws;
    float*    P      = dinv + n;
    float*    T      = P + (size_t)n * FDIM;
    unsigned* counts = (unsigned*)(T + (size_t)n * FDIM);
    unsigned* rowptr = counts + n;
    unsigned* cursor = rowptr + (n + 1);
    int*      colidx = (int*)(cursor + n);

    const long long* srcIdx = ei;
    const long long* dstIdx = ei + e;

    const int nb  = (n + 255) / 256;
    const int eb  = (e + 255) / 256;
    const int rb  = (n + 15) / 16;       // 16-row GEMM blocks
    const int ab  = (n + 7) / 8;         // 8 waves (nodes) per block

    // CSR by destination (integer atomics only)
    zero_counts_kernel<<<nb, 256, 0, stream>>>(counts, n);
    count_edges_kernel<<<eb, 256, 0, stream>>>(dstIdx, counts, e);
    scan_kernel<<<1, 256, 0, stream>>>(counts, rowptr, n);
    prep_kernel<<<nb, 256, 0, stream>>>(rowptr, counts, cursor, dinv, n);
    fill_csr_kernel<<<eb, 256, 0, stream>>>(srcIdx, dstIdx, cursor, colidx, e);

    // out = (1-alpha) * (k_node_pred @ kW + kb)
    gemm_wmma_kernel<64, false><<<rb, 256, 0, stream>>>(
        Kp, kW, nullptr, kb, alpha, out, n);

    // layer 1: P = (X @ W1) * dinv ; T = relu(dinv * (P[i] + sum_in P) + b1)
    gemm_wmma_kernel<128, true><<<rb, 256, 0, stream>>>(
        X, W1, dinv, nullptr, nullptr, P, n);
    aggregate_kernel<true><<<ab, 256, 0, stream>>>(
        P, rowptr, colidx, dinv, b1, nullptr, T, n);

    // layer 2: P = (T @ W2) * dinv ; out += alpha * (dinv * (P[i] + sum_in P) + b2)
    gemm_wmma_kernel<128, true><<<rb, 256, 0, stream>>>(
        T, W2, dinv, nullptr, nullptr, P, n);
    aggregate_kernel<false><<<ab, 256, 0, stream>>>(
        P, rowptr, colidx, dinv, b2, alpha, out, n);
}